// TemporalHyperbolicGNN_78606491451779
// MI455X (gfx1250) — compile-verified
//
#include <hip/hip_runtime.h>
#include <hip/hip_bf16.h>

typedef float v2f __attribute__((ext_vector_type(2)));
typedef float v8f __attribute__((ext_vector_type(8)));

// ---------------------------------------------------------------------------
// Zero a float buffer (workspace must be re-initialized every call).
// ---------------------------------------------------------------------------
__global__ __launch_bounds__(256) void thg_zero_kernel(float* __restrict__ p, long n) {
    long i = (long)blockIdx.x * blockDim.x + threadIdx.x;
    if (i < n) p[i] = 0.0f;
}

// ---------------------------------------------------------------------------
// Edge scatter: one wave32 per edge. Lane l owns features [2l, 2l+1] of the
// 64-wide source row; accumulate into sums[(t*N+dst)*64 + f] with f32 atomics.
// Lane 0 counts the edge.
// ---------------------------------------------------------------------------
__global__ __launch_bounds__(256) void thg_scatter_kernel(
    const float* __restrict__ X,     // [N, 64]
    const int*   __restrict__ src,   // [E]
    const int*   __restrict__ dst,   // [E]
    const int*   __restrict__ tidx,  // [E]
    float* __restrict__ sums,        // [T*N, 64]
    float* __restrict__ cnt,         // [T*N]
    int E, int N)
{
    int w    = (int)((blockIdx.x * (long)blockDim.x + threadIdx.x) >> 5);
    int lane = threadIdx.x & 31;
    if (w >= E) return;

    int  s = src[w];
    int  d = dst[w];
    int  t = tidx[w];
    long seg = (long)t * N + d;

    v2f xv = *(const v2f*)(X + (long)s * 64 + lane * 2);
    float* p = sums + seg * 64 + lane * 2;
    atomicAdd(p + 0, xv.x);
    atomicAdd(p + 1, xv.y);
    if (lane == 0) atomicAdd(cnt + seg, 1.0f);
}

// ---------------------------------------------------------------------------
// Mean + hyperbolic normalization, writing the [N, T*64] concatenated layout.
// One wave32 per segment row r = t*N + node. Lane l holds features 2l,2l+1.
//   mean = sums / max(cnt,1);   h = mean / (1 - c^2 * ||mean||^2)
// ---------------------------------------------------------------------------
__global__ __launch_bounds__(256) void thg_mean_norm_kernel(
    const float* __restrict__ sums,  // [T*N, 64]
    const float* __restrict__ cnt,   // [T*N]
    float* __restrict__ H,           // [N, T*64]
    int N, int T, float c)
{
    long R    = (long)T * N;
    long w    = (blockIdx.x * (long)blockDim.x + threadIdx.x) >> 5;
    int  lane = threadIdx.x & 31;
    if (w >= R) return;

    float cn  = fmaxf(cnt[w], 1.0f);
    float inv = 1.0f / cn;

    v2f sv = *(const v2f*)(sums + w * 64 + lane * 2);
    float mx = sv.x * inv;
    float my = sv.y * inv;

    // wave32 sum-of-squares reduction
    float ss = mx * mx + my * my;
    #pragma unroll
    for (int off = 16; off > 0; off >>= 1)
        ss += __shfl_xor(ss, off, 32);

    float r = 1.0f / (1.0f - (c * c) * ss);

    int t    = (int)(w / N);
    int node = (int)(w % N);
    v2f hv; hv.x = mx * r; hv.y = my * r;
    *(v2f*)(H + (long)node * (T * 64) + t * 64 + lane * 2) = hv;
}

// ---------------------------------------------------------------------------
// GEMM + bias + ReLU via V_WMMA_F32_16X16X4_F32.
// One wave32 per 16x16 output tile; K loop over KT in steps of 4.
// NC and KT are compile-time so every load in the unrolled loop is a
// base-register + immediate-offset global load (no per-iteration addr math).
//
// f32 fragment layouts (CDNA5 ISA):
//   A 16x4 : lane -> M=lane&15 ; VGPR{0,1} = K = 2*(lane>>4)+{0,1}
//   B 4x16 : lane -> N=lane&15 ; VGPR{0,1} = K = 2*(lane>>4)+{0,1}
//   C/D    : VGPR v, lane half h -> M = v + 8*h, N = lane&15
// ---------------------------------------------------------------------------
template <int NC, int KT>
__global__ __launch_bounds__(32) void thg_gemm_relu_wmma_kernel(
    const float* __restrict__ A,     // [M_total, KT]
    const float* __restrict__ W,     // [KT, NC]
    const float* __restrict__ bias,  // [NC]
    float* __restrict__ Y,           // [M_total, NC]
    int M_total)
{
    const int lane = threadIdx.x & 31;
    const int half = lane >> 4;      // 0 or 1
    const int lm   = lane & 15;

    const int row0 = blockIdx.x * 16;
    const int colg = blockIdx.y * 16 + lm;

    // Clamp A-row reads for a possible ragged last tile (EXEC must stay full).
    int arow = row0 + lm;
    if (arow > M_total - 1) arow = M_total - 1;
    const float* __restrict__ Arow = A + (long)arow * KT + half * 2;   // K=half*2 base
    const float* __restrict__ Wcol = W + (long)(half * 2) * NC + colg; // K=half*2 base

    v8f acc = {};
    #pragma unroll 8
    for (int k = 0; k < KT; k += 4) {
        v2f a = *(const v2f*)(Arow + k);              // A[M, k+half*2 .. +1]
        v2f b;
        b.x = Wcol[(long)k * NC];                     // W[k+half*2    , colg]
        b.y = Wcol[(long)k * NC + NC];                // W[k+half*2 + 1, colg]
        acc = __builtin_amdgcn_wmma_f32_16x16x4_f32(
            /*neg_a=*/false, a, /*neg_b=*/false, b,
            /*c_mod=*/(short)0, acc, /*reuse_a=*/false, /*reuse_b=*/false);
    }

    float bv = bias[colg];
    #pragma unroll
    for (int v = 0; v < 8; ++v) {
        int m = row0 + v + half * 8;
        if (m < M_total) {
            float val = acc[v] + bv;
            val = fmaxf(val, 0.0f);   // fused ReLU
            Y[(long)m * NC + colg] = val;
        }
    }
}

// ---------------------------------------------------------------------------
// Host orchestration
// ---------------------------------------------------------------------------
extern "C" void kernel_launch(void* const* d_in, const int* in_sizes, int n_in,
                              void* d_out, int out_size, void* d_ws, size_t ws_size,
                              hipStream_t stream) {
    const float* x     = (const float*)d_in[0];   // [N, 64]
    const int*   edge  = (const int*)  d_in[1];   // [2, E]
    const int*   tidx  = (const int*)  d_in[2];   // [E]
    const float* W1    = (const float*)d_in[3];   // [384, 64]
    const float* b1    = (const float*)d_in[4];   // [64]
    const float* W2    = (const float*)d_in[5];   // [384, 16]
    const float* b2    = (const float*)d_in[6];   // [16]
    float*       out   = (float*)d_out;           // [N, 16]

    constexpr int F   = 64;                       // features / hidden
    constexpr int T   = 6;                        // time steps (fixed by model)
    constexpr int KT  = T * F;                    // 384
    constexpr int NC2 = 16;                       // num classes
    const int N = in_sizes[0] / F;
    const int E = in_sizes[1] / 2;
    const float c = 0.1f;

    const int* src = edge;
    const int* dst = edge + E;

    const long TN = (long)T * N;

    // Workspace layout
    float* sums = (float*)d_ws;                   // [T*N, 64]
    float* cnt  = sums + TN * F;                  // [T*N]
    float* H    = cnt  + TN;                      // [N, 384]
    float* Y1   = H    + (long)N * KT;            // [N, 64]

    const long sumsN = TN * F;
    dim3 blk256(256);
    dim3 gZeroS((unsigned)((sumsN + 255) / 256));
    dim3 gZeroC((unsigned)((TN + 255) / 256));
    dim3 gScat((unsigned)((E + 7) / 8));          // 8 waves/block, 1 edge/wave
    dim3 gMean((unsigned)((TN + 7) / 8));
    dim3 blk32(32);
    dim3 gG1((unsigned)((N + 15) / 16), F   / 16); // 4 column tiles
    dim3 gG2((unsigned)((N + 15) / 16), NC2 / 16); // 1 column tile

    // -------- Layer 1 --------
    thg_zero_kernel<<<gZeroS, blk256, 0, stream>>>(sums, sumsN);
    thg_zero_kernel<<<gZeroC, blk256, 0, stream>>>(cnt, TN);
    thg_scatter_kernel<<<gScat, blk256, 0, stream>>>(x, src, dst, tidx, sums, cnt, E, N);
    thg_mean_norm_kernel<<<gMean, blk256, 0, stream>>>(sums, cnt, H, N, T, c);
    thg_gemm_relu_wmma_kernel<F, KT><<<gG1, blk32, 0, stream>>>(H, W1, b1, Y1, N);
    // (outer relu in the model is idempotent with the conv's relu)

    // -------- Layer 2 --------
    thg_zero_kernel<<<gZeroS, blk256, 0, stream>>>(sums, sumsN);
    thg_zero_kernel<<<gZeroC, blk256, 0, stream>>>(cnt, TN);
    thg_scatter_kernel<<<gScat, blk256, 0, stream>>>(Y1, src, dst, tidx, sums, cnt, E, N);
    thg_mean_norm_kernel<<<gMean, blk256, 0, stream>>>(sums, cnt, H, N, T, c);
    thg_gemm_relu_wmma_kernel<NC2, KT><<<gG2, blk32, 0, stream>>>(H, W2, b2, out, N);
}